// rnnmodel_11252814316201
// MI455X (gfx1250) — compile-verified
//
#include <hip/hip_runtime.h>
#include <math.h>

typedef __attribute__((ext_vector_type(2))) float v2f;
typedef __attribute__((ext_vector_type(8))) float v8f;

namespace {

constexpr int kB = 2048;  // batch
constexpr int kT = 512;   // time steps
constexpr int kH = 10;    // hidden size
constexpr int kL = 10;    // layers
constexpr int kS = 12;    // padded per-token stride in hidden activation buffers (48B -> 16B-aligned stores)

// D = A(16x4 f32) * B(4x16 f32) + C(16x16 f32), wave32.
__device__ __forceinline__ v8f wmma_k4(v2f a, v2f b, v8f c) {
  // 8 args: (neg_a, A, neg_b, B, c_mod, C, reuse_a, reuse_b)
  return __builtin_amdgcn_wmma_f32_16x16x4_f32(false, a, false, b, (short)0, c,
                                               false, false);
}

// lane <-> lane^16 swap. Prefer v_permlanex16_b32 (VALU pipe, no DS-counter
// round trip on the recurrence critical path); identity nibble selectors make
// each lane fetch its own index from the opposite 16-lane half.
__device__ __forceinline__ float swap16(float v) {
#if __has_builtin(__builtin_amdgcn_permlanex16)
  const int u = __float_as_int(v);
  const int r = __builtin_amdgcn_permlanex16(u, u, 0x76543210, 0xfedcba98,
                                             true, false);
  return __int_as_float(r);
#else
  return __shfl_xor(v, 16, 32);
#endif
}

// Branchless tanh on the recurrence critical path: CDNA5 hardware V_TANH_F32.
__device__ __forceinline__ float fast_tanh(float x) {
#if __has_builtin(__builtin_amdgcn_tanhf)
  return __builtin_amdgcn_tanhf(x);
#else
  const float e = __expf(2.0f * x);
  return 1.0f - 2.0f / (e + 1.0f);
#endif
}

// One RNN layer for this wave's 16 batch rows.
//   xin : [B, T, INS]  (reads features 0..F-1)
//   xout: [B, T, kS]   (writes hidden 0..kH-1)
// Transposed formulation: D(m=hid_out, n=batch) = W_hh*h^T + W_ih*x_t^T + bias
template <int F, int INS>
__device__ __forceinline__ void layer_pass(const float* __restrict__ xin,
                                           float* __restrict__ xout,
                                           const float* __restrict__ wih,
                                           const float* __restrict__ whh,
                                           const float* __restrict__ bih,
                                           const float* __restrict__ bhh,
                                           int lane, int b) {
  constexpr int KP = ((F + 3) / 4) * 4;  // padded K for input projection (4 or 12)
  constexpr int NP = KP / 4;             // input-proj WMMAs per step
  constexpr int NJ = KP / 2;             // B-layout regs for x_t^T
  const int half = lane >> 4;            // which half-wave
  const int mr = lane & 15;

  // ---- A-layout weights (loaded once per layer): lane holds row M=mr,
  // VGPR j of pair p holds K = 4p + j + 2*half.
  v2f whhA[3];
#pragma unroll
  for (int p = 0; p < 3; ++p)
#pragma unroll
    for (int j = 0; j < 2; ++j) {
      const int k = 4 * p + j + 2 * half;
      whhA[p][j] = (mr < kH && k < kH) ? whh[mr * kH + k] : 0.f;
    }
  v2f wihA[NP];
#pragma unroll
  for (int p = 0; p < NP; ++p)
#pragma unroll
    for (int j = 0; j < 2; ++j) {
      const int k = 4 * p + j + 2 * half;
      wihA[p][j] = (mr < kH && k < F) ? wih[mr * F + k] : 0.f;
    }

  // ---- bias in C layout: VGPR r holds M = r + 8*half (broadcast across lanes)
  v8f cbias;
#pragma unroll
  for (int r = 0; r < 8; ++r) {
    const int m = r + 8 * half;
    cbias[r] = (m < kH) ? (bih[m] + bhh[m]) : 0.f;
  }

  // ---- x_t^T gather setup: B reg j wants (feature f = 4*(j/2) + (j%2) + 2*half,
  // batch = b). Address advances by INS floats per time step.
  int xidx[NJ];
  float xmask[NJ];
#pragma unroll
  for (int j = 0; j < NJ; ++j) {
    const int f = 4 * (j / 2) + (j % 2) + 2 * half;
    const bool ok = (f < F);
    xmask[j] = ok ? 1.f : 0.f;
    xidx[j] = b * kT * INS + (ok ? f : 0);  // clamped: always in-bounds
  }

  float hB[6] = {0.f, 0.f, 0.f, 0.f, 0.f, 0.f};  // h0 = 0
  float* op = xout + (size_t)b * kT * kS;

#pragma unroll 2
  for (int t = 0; t < kT; ++t) {
    // gather x_t^T into B layout (independent of h -> overlaps the h-chain)
    float xB[NJ];
#pragma unroll
    for (int j = 0; j < NJ; ++j) {
      xB[j] = xmask[j] * xin[xidx[j]];
      xidx[j] += INS;
    }

    v8f acc = cbias;
#pragma unroll
    for (int p = 0; p < NP; ++p) {  // + W_ih * x_t^T (off the h critical path)
      v2f bx = {xB[2 * p], xB[2 * p + 1]};
      acc = wmma_k4(wihA[p], bx, acc);
    }
#pragma unroll
    for (int p = 0; p < 3; ++p) {   // + W_hh * h^T  (K = 12, top 2 padded 0)
      v2f bh = {hB[2 * p], hB[2 * p + 1]};
      acc = wmma_k4(whhA[p], bh, acc);
    }
#pragma unroll
    for (int r = 0; r < 8; ++r) acc[r] = fast_tanh(acc[r]);  // rows >=10 stay tanh(0)=0

    // ---- store h (D layout: lane<16 has m=0..7 in VGPR0..7; lane>=16 has m=8,9 in VGPR0,1)
    if (half == 0) {
      *reinterpret_cast<float4*>(op) = make_float4(acc[0], acc[1], acc[2], acc[3]);
      *reinterpret_cast<float4*>(op + 4) = make_float4(acc[4], acc[5], acc[6], acc[7]);
    } else {
      *reinterpret_cast<float2*>(op + 8) = make_float2(acc[0], acc[1]);
    }
    op += kS;

    // ---- rebuild h^T in B layout for the next step (6 VALU half-swaps + selects)
    const float s0 = swap16(acc[0]);  // m=8  values into lanes<16
    const float s1 = swap16(acc[1]);  // m=9
    const float s2 = swap16(acc[2]);  // m=2  values into lanes>=16
    const float s3 = swap16(acc[3]);  // m=3
    const float s6 = swap16(acc[6]);  // m=6
    const float s7 = swap16(acc[7]);  // m=7
    hB[0] = half ? s2 : acc[0];  // K = 0 | 2
    hB[1] = half ? s3 : acc[1];  // K = 1 | 3
    hB[2] = half ? s6 : acc[4];  // K = 4 | 6
    hB[3] = half ? s7 : acc[5];  // K = 5 | 7
    hB[4] = half ? 0.f : s0;     // K = 8 | 10(pad)
    hB[5] = half ? 0.f : s1;     // K = 9 | 11(pad)
  }
}

// One wave owns 16 batch rows and runs them through all 10 layers (no cross-batch
// mixing in an RNN -> no global sync needed; store->load same addr same wave is
// kept in order by hardware).
__global__ __launch_bounds__(256) void rnn_all_layers(
    const float* __restrict__ data, float* buf0, float* buf1,
    const float* __restrict__ wih0, const float* __restrict__ wihs,
    const float* __restrict__ whhs, const float* __restrict__ bihs,
    const float* __restrict__ bhhs) {
  const int lane = threadIdx.x & 31;
  const int wv = (blockIdx.x * blockDim.x + threadIdx.x) >> 5;
  const int b = wv * 16 + (lane & 15);

  layer_pass<3, 3>(data, buf0, wih0, whhs, bihs, bhhs, lane, b);

  const float* cur = buf0;
  float* nxt = buf1;
#pragma unroll 1
  for (int l = 1; l < kL; ++l) {
    layer_pass<kH, kS>(cur, nxt, wihs + (l - 1) * kH * kH, whhs + l * kH * kH,
                       bihs + l * kH, bhhs + l * kH, lane, b);
    const float* rd = cur;
    cur = nxt;
    nxt = const_cast<float*>(rd);
  }
  // after l=9 the result lives in buf1 (odd number of hidden layers -> buf1)
}

__global__ __launch_bounds__(256) void final_linear(const float* __restrict__ xl,
                                                    const float* __restrict__ lw,
                                                    const float* __restrict__ lb,
                                                    float* __restrict__ out) {
  const int i = blockIdx.x * blockDim.x + threadIdx.x;
  if (i >= kB * kT) return;
  const float* p = xl + (size_t)i * kS;
  float s = lb[0];
#pragma unroll
  for (int m = 0; m < kH; ++m) s = fmaf(p[m], lw[m], s);
  out[i] = s;
}

}  // namespace

extern "C" void kernel_launch(void* const* d_in, const int* in_sizes, int n_in,
                              void* d_out, int out_size, void* d_ws, size_t ws_size,
                              hipStream_t stream) {
  (void)in_sizes; (void)n_in; (void)out_size; (void)ws_size;
  const float* data = (const float*)d_in[0];  // [B,T,3]
  const float* wih0 = (const float*)d_in[1];  // [10,3]
  const float* wihs = (const float*)d_in[2];  // [9,10,10]
  const float* whhs = (const float*)d_in[3];  // [10,10,10]
  const float* bihs = (const float*)d_in[4];  // [10,10]
  const float* bhhs = (const float*)d_in[5];  // [10,10]
  const float* lw   = (const float*)d_in[6];  // [1,10]
  const float* lb   = (const float*)d_in[7];  // [1]

  // two ping-pong activation slabs [B,T,12] f32 (~50.3 MB each, L2-resident)
  float* buf0 = (float*)d_ws;
  float* buf1 = buf0 + (size_t)kB * kT * kS;

  // 128 waves total (one 16-row batch tile each): 16 blocks x 256 threads
  rnn_all_layers<<<16, 256, 0, stream>>>(data, buf0, buf1, wih0, wihs, whhs,
                                         bihs, bhhs);
  final_linear<<<(kB * kT + 255) / 256, 256, 0, stream>>>(buf1, lw, lb,
                                                          (float*)d_out);
}